// Pain_2628519985362
// MI455X (gfx1250) — compile-verified
//
#include <hip/hip_runtime.h>
#include <math.h>

#define F 128
#define NRBF 20
#define CUTOFF 5.0f
#define PI_F 3.14159265358979323846f
#define LSTR 132   // LDS row stride (floats): bank = (4*row + col) % 64 -> conflict-free

typedef __attribute__((ext_vector_type(2))) float v2f;
typedef __attribute__((ext_vector_type(8))) float v8f;

// ---------------- init: out = node_vec / node_s (accumulated into by edge kernel) --------
__global__ void init_out_kernel(const float* __restrict__ node_s,
                                const float* __restrict__ node_vec,
                                float* __restrict__ out_vec,
                                float* __restrict__ out_s,
                                int n_vec, int n_s) {
  int i = blockIdx.x * blockDim.x + threadIdx.x;
  int stride = gridDim.x * blockDim.x;
  for (int j = i; j < n_vec; j += stride) out_vec[j] = node_vec[j];
  for (int j = i; j < n_s;  j += stride) out_s[j]  = node_s[j];
}

// ---------------- filt = rbf @ Wf + bf (only first N rows are ever gathered), + ccut -----
// One wave per 16-row tile. 16 output tiles of 16 cols covering cols [0,128) U [256,384),
// stored compacted into filtC[Np][256] (Np = N rounded up to 16; padding rows are
// scratch-only, never read). K=20 -> 5 WMMA k-steps of 4. No store guards needed.
__global__ void filt_kernel(const float* __restrict__ edge_dis,
                            const float* __restrict__ Wf,   // [20][384]
                            const float* __restrict__ bf,   // [384]
                            float* __restrict__ filtC,      // [Np][256] compact
                            float* __restrict__ ccut,       // [Np]
                            int N) {
  __shared__ float sR[16 * NRBF];
  const int lane = threadIdx.x;       // one wave of 32
  const int i0 = blockIdx.x * 16;

  for (int idx = lane; idx < 16 * NRBF; idx += 32) {
    int r = idx / NRBF, k = idx % NRBF;
    int row = i0 + r; if (row >= N) row = N - 1;       // clamp (branch-free)
    float d = edge_dis[row];
    sR[idx] = __sinf((float)(k + 1) * (PI_F / CUTOFF) * d) / d;
  }
  if (lane < 16) {
    int row = i0 + lane;
    int crow = row >= N ? N - 1 : row;
    float d = edge_dis[crow];
    ccut[row] = (d <= CUTOFF) ? 0.5f * (__cosf(PI_F * d / CUTOFF) + 1.0f) : 0.0f; // pad rows ok
  }
  __syncthreads();

  const int r    = lane & 15;
  const int koff = (lane >> 4) * 2;
  const int mh   = (lane >> 4) * 8;
  for (int t = 0; t < 16; ++t) {
    const int nact = (t < 8 ? t * 16 : t * 16 + 128) + r;  // true column in [0,384)
    const int nsto = t * 16 + r;                           // compact column [0,256)
    float bfv = bf[nact];
    v8f c;
    #pragma unroll
    for (int j = 0; j < 8; ++j) c[j] = bfv;
    #pragma unroll
    for (int ks = 0; ks < 5; ++ks) {
      int k0 = ks * 4 + koff;
      v2f a, b;
      a.x = sR[r * NRBF + k0];
      a.y = sR[r * NRBF + k0 + 1];
      b.x = Wf[k0 * 384 + nact];
      b.y = Wf[(k0 + 1) * 384 + nact];
      c = __builtin_amdgcn_wmma_f32_16x16x4_f32(false, a, false, b, (short)0, c, false, false);
    }
    #pragma unroll
    for (int j = 0; j < 8; ++j)
      filtC[(i0 + j + mh) * 256 + nsto] = c[j];            // unguarded: padded buffer
  }
}

// ---------------- s_out = silu(node_s@W1+b1) @ W2 + b2, cols [0,128)U(256,384) -----------
// 8 waves / block, 16 node-rows per block. GEMM1: each wave owns one 16-col tile of the
// 128-wide h; h kept in LDS. GEMM2: each wave does 2 of the 16 needed col tiles.
__global__ void node_gemm_kernel(const float* __restrict__ node_s,
                                 const float* __restrict__ W1,  // [128][128]
                                 const float* __restrict__ b1,  // [128]
                                 const float* __restrict__ W2,  // [128][384]
                                 const float* __restrict__ b2,  // [384]
                                 float* __restrict__ soutC,     // [Np][256] compact
                                 int N) {
  __shared__ float sA[16 * LSTR];
  __shared__ float sH[16 * LSTR];
  const int tid  = threadIdx.x;      // 0..255
  const int wave = tid >> 5;
  const int lane = tid & 31;
  const int i0   = blockIdx.x * 16;

  for (int idx = tid; idx < 16 * F; idx += 256) {
    int row = idx >> 7, col = idx & 127;
    int grow = i0 + row; if (grow >= N) grow = N - 1;    // clamp (branch-free)
    sA[row * LSTR + col] = node_s[grow * F + col];
  }
  __syncthreads();

  const int r    = lane & 15;
  const int koff = (lane >> 4) * 2;
  const int mh   = (lane >> 4) * 8;

  { // GEMM1 + silu -> sH
    const int nc = wave * 16 + r;
    float b1v = b1[nc];
    v8f c;
    #pragma unroll
    for (int j = 0; j < 8; ++j) c[j] = b1v;
    #pragma unroll 4
    for (int ks = 0; ks < 32; ++ks) {
      int k0 = ks * 4 + koff;
      v2f a, b;
      a.x = sA[r * LSTR + k0];
      a.y = sA[r * LSTR + k0 + 1];
      b.x = W1[k0 * F + nc];
      b.y = W1[(k0 + 1) * F + nc];
      c = __builtin_amdgcn_wmma_f32_16x16x4_f32(false, a, false, b, (short)0, c, false, false);
    }
    #pragma unroll
    for (int j = 0; j < 8; ++j) {
      float x = c[j];
      sH[(j + mh) * LSTR + nc] = x / (1.0f + __expf(-x));  // silu
    }
  }
  __syncthreads();

  #pragma unroll
  for (int tt = 0; tt < 2; ++tt) { // GEMM2
    const int t = wave + tt * 8;
    const int nact = (t < 8 ? t * 16 : t * 16 + 128) + r;
    const int nsto = t * 16 + r;
    float b2v = b2[nact];
    v8f c;
    #pragma unroll
    for (int j = 0; j < 8; ++j) c[j] = b2v;
    #pragma unroll 4
    for (int ks = 0; ks < 32; ++ks) {
      int k0 = ks * 4 + koff;
      v2f a, b;
      a.x = sH[r * LSTR + k0];
      a.y = sH[r * LSTR + k0 + 1];
      b.x = W2[k0 * 384 + nact];
      b.y = W2[(k0 + 1) * 384 + nact];
      c = __builtin_amdgcn_wmma_f32_16x16x4_f32(false, a, false, b, (short)0, c, false, false);
    }
    #pragma unroll
    for (int j = 0; j < 8; ++j)
      soutC[(i0 + j + mh) * 256 + nsto] = c[j];            // unguarded: padded buffer
  }
}

// ---------------- edge phase: gather by src, scatter-add by dst (L2 resident) ------------
__global__ void edge_kernel(const int* __restrict__ edge,
                            const float* __restrict__ node_vec,
                            const float* __restrict__ filtC,
                            const float* __restrict__ soutC,
                            const float* __restrict__ ccut,
                            float* __restrict__ out_vec,
                            float* __restrict__ out_s,
                            int E_) {
  const int e = blockIdx.x * 2 + (threadIdx.x >> 7);
  const int f = threadIdx.x & 127;
  if (e >= E_) return;
  const int s = edge[2 * e + 1];   // src
  const int d = edge[2 * e + 0];   // dst
  const float cc  = ccut[s];
  const float fw0 = filtC[s * 256 + f]       * cc;   // gate_state filter
  const float fw2 = filtC[s * 256 + 128 + f] * cc;   // message_scalar filter
  const float so0 = soutC[s * 256 + f];
  const float so2 = soutC[s * 256 + 128 + f];
  const float gs = fw0 * so0;          // gate_state
  const float ms = fw2 * so2;          // message_scalar
  atomicAdd(&out_s[d * F + f], ms);
  const float v0 = node_vec[(s * F + f) * 3 + 0];
  const float v1 = node_vec[(s * F + f) * 3 + 1];
  const float v2 = node_vec[(s * F + f) * 3 + 2];
  atomicAdd(&out_vec[(d * F + f) * 3 + 0], v0 * gs);
  atomicAdd(&out_vec[(d * F + f) * 3 + 1], v1 * gs);
  atomicAdd(&out_vec[(d * F + f) * 3 + 2], v2 * gs);
}

extern "C" void kernel_launch(void* const* d_in, const int* in_sizes, int n_in,
                              void* d_out, int out_size, void* d_ws, size_t ws_size,
                              hipStream_t stream) {
  const float* node_s   = (const float*)d_in[0];
  const float* node_vec = (const float*)d_in[1];
  const int*   edge     = (const int*)d_in[2];
  /* d_in[3] edge_difference: unused by the reference math */
  const float* edge_dis = (const float*)d_in[4];
  const float* W1 = (const float*)d_in[5];
  const float* b1 = (const float*)d_in[6];
  const float* W2 = (const float*)d_in[7];
  const float* b2 = (const float*)d_in[8];
  const float* Wf = (const float*)d_in[9];
  const float* bf = (const float*)d_in[10];

  const int N  = in_sizes[0] / F;
  const int E_ = in_sizes[2] / 2;
  const int Np = (N + 15) & ~15;        // padded row count: stores never need guards

  float* out_vec = (float*)d_out;                       // [N][F][3]
  float* out_s   = out_vec + (size_t)N * F * 3;         // [N][F]

  float* filtC = (float*)d_ws;                          // [Np][256]
  float* soutC = filtC + (size_t)Np * 256;              // [Np][256]
  float* ccut  = soutC + (size_t)Np * 256;              // [Np]

  init_out_kernel<<<1024, 256, 0, stream>>>(node_s, node_vec, out_vec, out_s,
                                            N * F * 3, N * F);
  filt_kernel<<<Np / 16, 32, 0, stream>>>(edge_dis, Wf, bf, filtC, ccut, N);
  node_gemm_kernel<<<Np / 16, 256, 0, stream>>>(node_s, W1, b1, W2, b2, soutC, N);
  edge_kernel<<<(E_ + 1) / 2, 256, 0, stream>>>(edge, node_vec, filtC, soutC, ccut,
                                                out_vec, out_s, E_);
}